// EMGNN_16716012716348
// MI455X (gfx1250) — compile-verified
//
#include <hip/hip_runtime.h>

#define ALPHA 0.2f

typedef __attribute__((ext_vector_type(2))) float v2f;
typedef __attribute__((ext_vector_type(8))) float v8f;

__device__ __forceinline__ float leaky_f(float v) { return v > 0.f ? v : ALPHA * v; }

// ---------------- fill ----------------
__global__ void emgnn_fill_kernel(float* __restrict__ p, float v, long n) {
  long i = (long)blockIdx.x * blockDim.x + threadIdx.x;
  if (i < n) p[i] = v;
}

// ---------------- degree count (atomic) ----------------
__global__ void emgnn_deg_kernel(float* __restrict__ deg, const int* __restrict__ dst, long nE) {
  long i = (long)blockIdx.x * blockDim.x + threadIdx.x;
  if (i < nE) atomicAdd(&deg[dst[i]], 1.0f);
}

// ---------------- in-place deg -> dis = rsqrt(deg) ----------------
__global__ void emgnn_rsqrt_kernel(float* __restrict__ p, long n) {
  long i = (long)blockIdx.x * blockDim.x + threadIdx.x;
  if (i < n) {
    float d = p[i];
    p[i] = d > 0.f ? rsqrtf(d) : 0.f;
  }
}

// ---------------- GEMM: Y[rows x 64] = act(X[rows x K] @ W[K x 64] + b) ----------------
// Block = 128 threads (4 waves). Block tile = 16 rows x 64 cols; wave w owns cols [16w,16w+16).
// Uses V_WMMA_F32_16X16X4_F32, accumulating over K in steps of 4.
__global__ __launch_bounds__(128) void emgnn_gemm64_kernel(
    const float* __restrict__ X, const float* __restrict__ W,
    const float* __restrict__ bias, float* __restrict__ Y,
    int rows, int K, int act) {
  __shared__ float sX[16 * 128];   // X tile (K <= 128)
  __shared__ float sW[128 * 64];   // full W (K <= 128)

  const int row0 = blockIdx.x * 16;

  // stage X tile (zero-pad rows past end so the WMMA loop needs no guards)
  for (int i = threadIdx.x; i < 16 * K; i += blockDim.x) {
    int rr = i / K, cc = i - rr * K;
    int gr = row0 + rr;
    sX[i] = (gr < rows) ? X[(long)gr * K + cc] : 0.f;
  }
  for (int i = threadIdx.x; i < K * 64; i += blockDim.x) sW[i] = W[i];
  __syncthreads();

  const int lane = threadIdx.x & 31;
  const int wv   = threadIdx.x >> 5;   // 0..3
  const int col0 = wv * 16;
  const int m    = lane & 15;
  const int hi   = lane >> 4;          // 0: K pair {k,k+1}, 1: {k+2,k+3}

  v8f acc = {};
  for (int k = 0; k < K; k += 4) {
    int kk = k + 2 * hi;
    v2f a, b;
    // A 16x4 layout: lane m holds A[m][kk] in v0, A[m][kk+1] in v1
    a.x = sX[m * K + kk];
    a.y = sX[m * K + kk + 1];
    // B 4x16 layout: VGPR0 = rows {kk}, VGPR1 = rows {kk+1}, N = lane%16
    b.x = sW[kk * 64 + col0 + m];
    b.y = sW[(kk + 1) * 64 + col0 + m];
    acc = __builtin_amdgcn_wmma_f32_16x16x4_f32(
        /*neg_a=*/false, a, /*neg_b=*/false, b,
        /*c_mod=*/(short)0, acc, /*reuse_a=*/false, /*reuse_b=*/false);
  }

  float bb = bias ? bias[col0 + m] : 0.f;
#pragma unroll
  for (int r = 0; r < 8; ++r) {
    int orow = row0 + r + 8 * hi;   // C/D layout: VGPR r = rows r (lanes 0-15), r+8 (lanes 16-31)
    if (orow < rows) {
      float v = acc[r] + bb;
      if (act) v = leaky_f(v);
      Y[(long)orow * 64 + col0 + m] = v;
    }
  }
}

// ---------------- self-loop init: a[i][j] = t[i][j] * dis[i]^2 ----------------
__global__ void emgnn_selfinit_kernel(float* __restrict__ a, const float* __restrict__ t,
                                      const float* __restrict__ dis, long n64) {
  long i = (long)blockIdx.x * blockDim.x + threadIdx.x;
  if (i < n64) {
    float d = dis[i >> 6];
    a[i] = t[i] * d * d;
  }
}

// ---------------- edge scatter: one wave32 per edge, 2 floats per lane ----------------
__global__ void emgnn_scatter_kernel(float* __restrict__ a, const float* __restrict__ t,
                                     const int* __restrict__ src, const int* __restrict__ dst,
                                     const float* __restrict__ dis, long nE) {
  long gid = (long)blockIdx.x * blockDim.x + threadIdx.x;
  long e = gid >> 5;
  int lane = (int)(gid & 31);
  if (e >= nE) return;
  int s = src[e], d = dst[e];
  float w = dis[s] * dis[d];
  float v0 = t[(long)s * 64 + lane] * w;
  float v1 = t[(long)s * 64 + 32 + lane] * w;
  atomicAdd(&a[(long)d * 64 + lane], v0);
  atomicAdd(&a[(long)d * 64 + 32 + lane], v1);
}

// ---------------- y = leaky(a + b[col]) ----------------
__global__ void emgnn_bias_leaky_kernel(float* __restrict__ y, const float* __restrict__ a,
                                        const float* __restrict__ b, long n64) {
  long i = (long)blockIdx.x * blockDim.x + threadIdx.x;
  if (i < n64) y[i] = leaky_f(a[i] + b[i & 63]);
}

// ---------------- classifier (H=64 -> C=2) + log_softmax ----------------
__global__ void emgnn_cls_kernel(const float* __restrict__ z, const float* __restrict__ Wc,
                                 const float* __restrict__ bc, float* __restrict__ out, int rows) {
  int r = blockIdx.x * blockDim.x + threadIdx.x;
  if (r >= rows) return;
  const float* zr = z + (long)r * 64;
  float a0 = bc[0], a1 = bc[1];
#pragma unroll 8
  for (int j = 0; j < 64; ++j) {
    float v = zr[j];
    a0 += v * Wc[2 * j];
    a1 += v * Wc[2 * j + 1];
  }
  float mx = fmaxf(a0, a1);
  float lse = mx + logf(expf(a0 - mx) + expf(a1 - mx));
  out[2 * r]     = a0 - lse;
  out[2 * r + 1] = a1 - lse;
}

static inline unsigned cdivu(long a, long b) { return (unsigned)((a + b - 1) / b); }

extern "C" void kernel_launch(void* const* d_in, const int* in_sizes, int n_in,
                              void* d_out, int out_size, void* d_ws, size_t ws_size,
                              hipStream_t stream) {
  const float* x      = (const float*)d_in[0];
  const int*   ei     = (const int*)d_in[1];     // [2, E] row-major
  const float* meta_x = (const float*)d_in[2];
  const int*   mei    = (const int*)d_in[3];     // [2, ME]
  const float* W_lin  = (const float*)d_in[4];
  const float* b_lin  = (const float*)d_in[5];
  const float* W_mlin = (const float*)d_in[6];
  const float* b_mlin = (const float*)d_in[7];
  const float* W0     = (const float*)d_in[8];
  const float* b0     = (const float*)d_in[9];
  const float* W1     = (const float*)d_in[10];
  const float* b1     = (const float*)d_in[11];
  const float* Wm     = (const float*)d_in[12];
  const float* bm     = (const float*)d_in[13];
  const float* Wc     = (const float*)d_in[14];
  const float* bc     = (const float*)d_in[15];

  const int  F  = 128;
  const int  N  = in_sizes[0] / F;
  const long E  = in_sizes[1] / 2;
  const int  M  = in_sizes[2] / F;
  const long ME = in_sizes[3] / 2;
  const int  NM = N + M;

  const int* src  = ei;
  const int* dst  = ei + E;
  const int* msrc = mei;
  const int* mdst = mei + ME;

  // workspace carve-out
  char* ws = (char*)d_ws;
  size_t off = 0;
  auto carve = [&](size_t bytes) -> float* {
    float* p = (float*)(ws + off);
    off = (off + bytes + 255) & ~(size_t)255;
    return p;
  };
  float* h        = carve((size_t)NM * 64 * 4);  // node features; rows [N,N+M) hold mh, later z
  float* t        = carve((size_t)NM * 64 * 4);  // GEMM temp
  float* a        = carve((size_t)NM * 64 * 4);  // aggregation accumulator
  float* dis_data = carve((size_t)N * 4);
  float* dis_meta = carve((size_t)NM * 4);

  const int TPB = 256;
  const long n64N  = (long)N * 64;
  const long n64NM = (long)NM * 64;

  // ---- normalization coefficients ----
  // data graph (add_loops=True): deg = 1 (self) + incoming edge count
  emgnn_fill_kernel<<<cdivu(N, TPB), TPB, 0, stream>>>(dis_data, 1.0f, N);
  emgnn_deg_kernel<<<cdivu(E, TPB), TPB, 0, stream>>>(dis_data, dst, E);
  emgnn_rsqrt_kernel<<<cdivu(N, TPB), TPB, 0, stream>>>(dis_data, N);
  // meta graph (loops already included in edge list)
  emgnn_fill_kernel<<<cdivu(NM, TPB), TPB, 0, stream>>>(dis_meta, 0.0f, NM);
  emgnn_deg_kernel<<<cdivu(ME, TPB), TPB, 0, stream>>>(dis_meta, mdst, ME);
  emgnn_rsqrt_kernel<<<cdivu(NM, TPB), TPB, 0, stream>>>(dis_meta, NM);

  // ---- input projections (fused bias + leaky) ----
  emgnn_gemm64_kernel<<<cdivu(N, 16), 128, 0, stream>>>(x, W_lin, b_lin, h, N, F, 1);
  emgnn_gemm64_kernel<<<cdivu(M, 16), 128, 0, stream>>>(meta_x, W_mlin, b_mlin,
                                                        h + (long)N * 64, M, F, 1);

  // ---- GCN layer 0 (data graph) ----
  emgnn_gemm64_kernel<<<cdivu(N, 16), 128, 0, stream>>>(h, W0, nullptr, t, N, 64, 0);
  emgnn_selfinit_kernel<<<cdivu(n64N, TPB), TPB, 0, stream>>>(a, t, dis_data, n64N);
  emgnn_scatter_kernel<<<cdivu(E * 32, TPB), TPB, 0, stream>>>(a, t, src, dst, dis_data, E);
  emgnn_bias_leaky_kernel<<<cdivu(n64N, TPB), TPB, 0, stream>>>(h, a, b0, n64N);

  // ---- GCN layer 1 (data graph) ----
  emgnn_gemm64_kernel<<<cdivu(N, 16), 128, 0, stream>>>(h, W1, nullptr, t, N, 64, 0);
  emgnn_selfinit_kernel<<<cdivu(n64N, TPB), TPB, 0, stream>>>(a, t, dis_data, n64N);
  emgnn_scatter_kernel<<<cdivu(E * 32, TPB), TPB, 0, stream>>>(a, t, src, dst, dis_data, E);
  emgnn_bias_leaky_kernel<<<cdivu(n64N, TPB), TPB, 0, stream>>>(h, a, b1, n64N);

  // ---- meta fusion layer over z = [h; mh], no added loops ----
  emgnn_gemm64_kernel<<<cdivu(NM, 16), 128, 0, stream>>>(h, Wm, nullptr, t, NM, 64, 0);
  emgnn_fill_kernel<<<cdivu(n64NM, TPB), TPB, 0, stream>>>(a, 0.0f, n64NM);
  emgnn_scatter_kernel<<<cdivu(ME * 32, TPB), TPB, 0, stream>>>(a, t, msrc, mdst, dis_meta, ME);
  emgnn_bias_leaky_kernel<<<cdivu(n64NM, TPB), TPB, 0, stream>>>(h, a, bm, n64NM);

  // ---- classifier + log_softmax ----
  emgnn_cls_kernel<<<cdivu(NM, TPB), TPB, 0, stream>>>(h, Wc, bc, (float*)d_out, NM);
}